// GAT_33818572488975
// MI455X (gfx1250) — compile-verified
//
#include <hip/hip_runtime.h>
#include <hip/hip_bf16.h>
#include <math.h>

#define NEG_SLOPE 0.2f

typedef __attribute__((ext_vector_type(2))) float v2f;
typedef __attribute__((ext_vector_type(8))) float v8f;

// ---------------------------------------------------------------------------
// float atomic max via signed/unsigned integer ordering trick (init = -inf)
// ---------------------------------------------------------------------------
__device__ __forceinline__ void atomicMaxF32(float* addr, float val) {
  if (val >= 0.0f) {
    atomicMax((int*)addr, __float_as_int(val));
  } else {
    atomicMin((unsigned int*)addr, __float_as_uint(val));
  }
}

__global__ void fill_f32(float* __restrict__ p, float v, unsigned int n) {
  unsigned int i = blockIdx.x * blockDim.x + threadIdx.x;
  if (i < n) p[i] = v;
}

// ---------------------------------------------------------------------------
// C[M x Ncol] = A[M x K] * B[K x Ncol], f32, via V_WMMA_F32_16X16X4_F32.
// One wave (32 lanes) per 16x16 output tile. M % 16 == 0, K % 4 == 0,
// Ncol % 16 == 0 (holds here: M=100000, K in {128,64}, Ncol in {64,32}).
// A frag (16x4): lanes 0-15 row M=l, VGPR0=K0 VGPR1=K1; lanes 16-31 K2/K3.
// B frag (4x16): symmetric (col N=l&15, K pair selected by lane half).
// C/D (16x16): VGPR v -> row v (lanes 0-15) / row v+8 (lanes 16-31).
// ---------------------------------------------------------------------------
__global__ __launch_bounds__(32) void gemm_f32_wmma(
    const float* __restrict__ A, const float* __restrict__ B,
    float* __restrict__ C, int K, int Ncol) {
  const int m0 = blockIdx.x * 16;
  const int n0 = blockIdx.y * 16;
  const int lane = threadIdx.x;
  const int half = lane >> 4;   // 0: K{0,1}, 1: K{2,3} of each 4-step
  const int l16  = lane & 15;

  v8f acc = {};
  const float* arow = A + (size_t)(m0 + l16) * K + 2 * half;
  const float* bcol = B + (size_t)(2 * half) * Ncol + (n0 + l16);

  for (int k0 = 0; k0 < K; k0 += 4) {
    v2f a = *(const v2f*)(arow + k0);          // 8B-aligned (k even, K even)
    v2f b;
    b.x = bcol[(size_t)k0 * Ncol];
    b.y = bcol[(size_t)k0 * Ncol + Ncol];
    acc = __builtin_amdgcn_wmma_f32_16x16x4_f32(
        /*neg_a=*/false, a, /*neg_b=*/false, b,
        /*c_mod=*/(short)0, acc, /*reuse_a=*/false, /*reuse_b=*/false);
  }

  float* crow = C + (size_t)(m0 + 8 * half) * Ncol + (n0 + l16);
#pragma unroll
  for (int v = 0; v < 8; ++v) crow[(size_t)v * Ncol] = acc[v];
}

// ---------------------------------------------------------------------------
// Per-node attention coefficients: als[n,h] = <h[n,h,:], a_src[h,:]>, same ald.
// h laid out [N, H*C]; thread i = n*H + h, row offset = i*C.
// ---------------------------------------------------------------------------
template <int H, int C>
__global__ void attn_coef(const float* __restrict__ hf,
                          const float* __restrict__ asrc,
                          const float* __restrict__ adst,
                          float* __restrict__ als, float* __restrict__ ald,
                          unsigned int NH) {
  unsigned int i = blockIdx.x * blockDim.x + threadIdx.x;
  if (i >= NH) return;
  unsigned int hd = i % H;
  const float* row = hf + (size_t)i * C;
  float s = 0.f, d = 0.f;
#pragma unroll
  for (int c = 0; c < C; ++c) {
    float v = row[c];
    s += v * asrc[hd * C + c];
    d += v * adst[hd * C + c];
  }
  als[i] = s;
  ald[i] = d;
}

__device__ __forceinline__ void edge_nodes(const long long* __restrict__ ei,
                                           unsigned int e, unsigned int E,
                                           unsigned int& src, unsigned int& dst) {
  if (e < E) {
    src = (unsigned int)ei[e];
    dst = (unsigned int)ei[(size_t)E + e];
  } else {                 // self loops appended after the E real edges
    src = dst = e - E;
  }
}

// Pass 1: per-(edge,head) leaky-relu logit -> segment max over dst
template <int H>
__global__ void edge_logit_max(const float* __restrict__ als,
                               const float* __restrict__ ald,
                               const long long* __restrict__ ei,
                               unsigned int E, unsigned int tot,
                               float* __restrict__ m) {
  unsigned int i = blockIdx.x * blockDim.x + threadIdx.x;
  if (i >= tot) return;
  unsigned int e = i / H;
  unsigned int h = i % H;
  unsigned int src, dst;
  edge_nodes(ei, e, E, src, dst);
  float l = als[src * H + h] + ald[dst * H + h];
  l = (l < 0.f) ? NEG_SLOPE * l : l;
  atomicMaxF32(&m[dst * H + h], l);
}

// Pass 2: e = exp(logit - m[dst]); store per-edge, segment-sum into z[dst]
template <int H>
__global__ void edge_exp_sum(const float* __restrict__ als,
                             const float* __restrict__ ald,
                             const float* __restrict__ m,
                             const long long* __restrict__ ei,
                             unsigned int E, unsigned int tot,
                             float* __restrict__ ex, float* __restrict__ z) {
  unsigned int i = blockIdx.x * blockDim.x + threadIdx.x;
  if (i >= tot) return;
  unsigned int e = i / H;
  unsigned int h = i % H;
  unsigned int src, dst;
  edge_nodes(ei, e, E, src, dst);
  float l = als[src * H + h] + ald[dst * H + h];
  l = (l < 0.f) ? NEG_SLOPE * l : l;
  float ev = expf(l - m[dst * H + h]);
  ex[i] = ev;
  atomicAdd(&z[dst * H + h], ev);
}

// Pass 2.5: normalize once per (edge, head) so the heavy pass doesn't re-gather z
template <int H>
__global__ void edge_norm(float* __restrict__ ex, const float* __restrict__ z,
                          const long long* __restrict__ ei,
                          unsigned int E, unsigned int tot) {
  unsigned int i = blockIdx.x * blockDim.x + threadIdx.x;
  if (i >= tot) return;
  unsigned int e = i / H;
  unsigned int h = i % H;
  unsigned int src, dst;
  edge_nodes(ei, e, E, src, dst);
  ex[i] = ex[i] / (z[dst * H + h] + 1e-16f);
}

// Pass 3: out[dst, h, c] += h[src, h, c] * alpha[e,h]
// channel-major thread mapping -> coalesced gather + coalesced atomics.
template <int H, int C>
__global__ void edge_aggregate(const float* __restrict__ hf,
                               const float* __restrict__ alpha,
                               const long long* __restrict__ ei,
                               unsigned int E, unsigned int tot,
                               float* __restrict__ out) {
  constexpr int HC = H * C;
  unsigned int i = blockIdx.x * blockDim.x + threadIdx.x;
  if (i >= tot) return;
  unsigned int e = i / HC;
  unsigned int j = i % HC;
  unsigned int h = j / C;
  unsigned int src, dst;
  edge_nodes(ei, e, E, src, dst);
  float a = alpha[e * H + h];
  atomicAdd(&out[dst * HC + j], hf[src * HC + j] * a);
}

// bias add, optional ELU (alpha = 1)
template <int D, bool ELU>
__global__ void bias_act(float* __restrict__ p, const float* __restrict__ b,
                         unsigned int n) {
  unsigned int i = blockIdx.x * blockDim.x + threadIdx.x;
  if (i >= n) return;
  float v = p[i] + b[i % D];
  if (ELU) v = (v > 0.f) ? v : (expf(v) - 1.0f);
  p[i] = v;
}

// ---------------------------------------------------------------------------

static inline unsigned int cdivu(unsigned int a, unsigned int b) {
  return (a + b - 1) / b;
}

extern "C" void kernel_launch(void* const* d_in, const int* in_sizes, int n_in,
                              void* d_out, int out_size, void* d_ws, size_t ws_size,
                              hipStream_t stream) {
  const float*     x     = (const float*)d_in[0];
  const long long* ei    = (const long long*)d_in[1];   // int64 [2, E]
  const float*     W1    = (const float*)d_in[2];       // [128, 64]
  const float*     asrc1 = (const float*)d_in[3];       // [4, 16]
  const float*     adst1 = (const float*)d_in[4];
  const float*     b1    = (const float*)d_in[5];       // [64]
  const float*     W2    = (const float*)d_in[6];       // [64, 32]
  const float*     asrc2 = (const float*)d_in[7];       // [1, 32]
  const float*     adst2 = (const float*)d_in[8];
  const float*     b2    = (const float*)d_in[9];       // [32]
  float* out = (float*)d_out;

  const unsigned int Nn = (unsigned int)(in_sizes[0] / 128);   // 100000
  const unsigned int E  = (unsigned int)(in_sizes[1] / 2);     // 1600000
  const unsigned int ET = E + Nn;                              // with self loops
  constexpr int H1 = 4, C1 = 16, D1 = 64, D2 = 32;

  // workspace layout (floats); total ~106 MB
  float* ws = (float*)d_ws;
  size_t off = 0;
  float* h1   = ws + off; off += (size_t)Nn * D1;
  float* als1 = ws + off; off += (size_t)Nn * H1;
  float* ald1 = ws + off; off += (size_t)Nn * H1;
  float* m1   = ws + off; off += (size_t)Nn * H1;
  float* z1   = ws + off; off += (size_t)Nn * H1;
  float* ex1  = ws + off; off += (size_t)ET * H1;
  float* o1   = ws + off; off += (size_t)Nn * D1;
  float* h2   = ws + off; off += (size_t)Nn * D2;
  float* als2 = ws + off; off += (size_t)Nn;
  float* ald2 = ws + off; off += (size_t)Nn;
  float* m2   = ws + off; off += (size_t)Nn;
  float* z2   = ws + off; off += (size_t)Nn;
  float* ex2  = ws + off; off += (size_t)ET;

  const unsigned int TB = 256;
  const float NEG_INF = -INFINITY;

  // ---- init accumulators (d_out & ws are poisoned/stale each run) ----
  fill_f32<<<cdivu(Nn * H1, TB), TB, 0, stream>>>(m1, NEG_INF, Nn * H1);
  fill_f32<<<cdivu(Nn * H1, TB), TB, 0, stream>>>(z1, 0.f, Nn * H1);
  fill_f32<<<cdivu(Nn * D1, TB), TB, 0, stream>>>(o1, 0.f, Nn * D1);
  fill_f32<<<cdivu(Nn, TB), TB, 0, stream>>>(m2, NEG_INF, Nn);
  fill_f32<<<cdivu(Nn, TB), TB, 0, stream>>>(z2, 0.f, Nn);
  fill_f32<<<cdivu(Nn * D2, TB), TB, 0, stream>>>(out, 0.f, Nn * D2);

  // ---- layer 1: GATConv(128 -> 4 heads x 16, concat) + ELU ----
  gemm_f32_wmma<<<dim3(Nn / 16, D1 / 16), 32, 0, stream>>>(x, W1, h1, 128, D1);
  attn_coef<H1, C1><<<cdivu(Nn * H1, TB), TB, 0, stream>>>(h1, asrc1, adst1,
                                                           als1, ald1, Nn * H1);
  edge_logit_max<H1><<<cdivu(ET * H1, TB), TB, 0, stream>>>(als1, ald1, ei, E,
                                                            ET * H1, m1);
  edge_exp_sum<H1><<<cdivu(ET * H1, TB), TB, 0, stream>>>(als1, ald1, m1, ei, E,
                                                          ET * H1, ex1, z1);
  edge_norm<H1><<<cdivu(ET * H1, TB), TB, 0, stream>>>(ex1, z1, ei, E, ET * H1);
  edge_aggregate<H1, C1><<<cdivu(ET * H1 * C1, TB), TB, 0, stream>>>(h1, ex1, ei, E,
                                                                     ET * H1 * C1, o1);
  bias_act<D1, true><<<cdivu(Nn * D1, TB), TB, 0, stream>>>(o1, b1, Nn * D1);

  // ---- layer 2: GATConv(64 -> 1 head x 32, mean==identity), no activation ----
  gemm_f32_wmma<<<dim3(Nn / 16, D2 / 16), 32, 0, stream>>>(o1, W2, h2, D1, D2);
  attn_coef<1, D2><<<cdivu(Nn, TB), TB, 0, stream>>>(h2, asrc2, adst2,
                                                     als2, ald2, Nn);
  edge_logit_max<1><<<cdivu(ET, TB), TB, 0, stream>>>(als2, ald2, ei, E, ET, m2);
  edge_exp_sum<1><<<cdivu(ET, TB), TB, 0, stream>>>(als2, ald2, m2, ei, E, ET,
                                                    ex2, z2);
  edge_norm<1><<<cdivu(ET, TB), TB, 0, stream>>>(ex2, z2, ei, E, ET);
  edge_aggregate<1, D2><<<cdivu(ET * D2, TB), TB, 0, stream>>>(h2, ex2, ei, E,
                                                               ET * D2, out);
  bias_act<D2, false><<<cdivu(Nn * D2, TB), TB, 0, stream>>>(out, b2, Nn * D2);
}